// PseudoLevinsonDurbinRecursion_13331578486936
// MI455X (gfx1250) — compile-verified
//
#include <hip/hip_runtime.h>
#include <hip/hip_bf16.h>
#include <stdint.h>

// ---------------------------------------------------------------------------
// PseudoLevinsonDurbinRecursion on MI455X (gfx1250)
//
// 65,536 independent order-32 symmetric-Toeplitz solves. Memory floor
// ~0.74us @ 23.3TB/s; Levinson-Durbin is 10x fewer FLOPs than the dense
// solve and maps to pure VALU FMA with compile-time indices. CDNA5-specific
// path used: async global->LDS DMA (ASYNCcnt) for coalesced staging.
// ---------------------------------------------------------------------------

#define SYS_PER_BLOCK 128
#define M_ORDER 32
#define RLEN 33   // r[0..32] per system

typedef __attribute__((address_space(3))) float lds_float_t;

__global__ __launch_bounds__(SYS_PER_BLOCK) void
pseudo_levinson_kernel(const float* __restrict__ r, float* __restrict__ out,
                       int nsys) {
    __shared__ __align__(16) float smem[SYS_PER_BLOCK * RLEN];

    const int tid = threadIdx.x;
    const long long blockBase = (long long)blockIdx.x * SYS_PER_BLOCK;
    int sysInBlock = (int)(nsys - blockBase);
    if (sysInBlock > SYS_PER_BLOCK) sysInBlock = SYS_PER_BLOCK;
    if (sysInBlock <= 0) return;

    // ---- Stage r for this block's systems into LDS via async DMA ----------
    const float* gbase = r + blockBase * RLEN;
    const uint32_t lds0 = (uint32_t)(uintptr_t)(lds_float_t*)&smem[0];
    const int nbytes = sysInBlock * RLEN * 4;

    if (((nbytes & 15) == 0) && ((((uintptr_t)gbase) & 15) == 0)) {
        // 16-byte chunks, fully coalesced (holds for full blocks: 128*33*4 = 1056*16)
        const int nchunk = nbytes >> 4;
        for (int i = tid; i < nchunk; i += SYS_PER_BLOCK) {
            uint32_t lds_addr = lds0 + ((uint32_t)i << 4);
            uint32_t goff     = (uint32_t)i << 4;
            asm volatile(
                "global_load_async_to_lds_b128 %0, %1, %2 offset:0"
                :
                : "v"(lds_addr), "v"(goff), "s"(gbase)
                : "memory");
        }
    } else {
        // dword fallback for ragged tails
        const int nd = nbytes >> 2;
        for (int i = tid; i < nd; i += SYS_PER_BLOCK) {
            uint32_t lds_addr = lds0 + ((uint32_t)i << 2);
            uint32_t goff     = (uint32_t)i << 2;
            asm volatile(
                "global_load_async_to_lds_b32 %0, %1, %2 offset:0"
                :
                : "v"(lds_addr), "v"(goff), "s"(gbase)
                : "memory");
        }
    }
    // Wait for this wave's async DMA, then make all waves' LDS writes visible.
    asm volatile("s_wait_asynccnt 0x0" ::: "memory");
    __syncthreads();

    if (tid < sysInBlock) {
        // ---- Pull my system's r[0..32] out of LDS (bank-conflict free:
        //      stride 33 is odd => 33*t mod 64 injective over a wave). -------
        float rr[RLEN];
        const int base = tid * RLEN;
#pragma unroll
        for (int j = 0; j < RLEN; ++j) rr[j] = smem[base + j];

        // ---- Fully-unrolled Levinson-Durbin (Golub & Van Loan 4.7.1):
        //      solves R a = -r[1..M], with E_M = r0 + r1.a maintained free. --
        float A[M_ORDER];
        float E = rr[0];
#pragma unroll
        for (int k = 0; k < M_ORDER; ++k) {
            float acc = rr[k + 1];
#pragma unroll
            for (int j = 0; j < k; ++j) acc = fmaf(A[j], rr[k - j], acc);

            const float kr = -acc * __builtin_amdgcn_rcpf(E);

            const int half = k >> 1;
#pragma unroll
            for (int j = 0; j < half; ++j) {
                const float t0 = A[j];
                const float t1 = A[k - 1 - j];
                A[j]         = fmaf(kr, t1, t0);
                A[k - 1 - j] = fmaf(kr, t0, t1);
            }
            if (k & 1) A[half] = fmaf(kr, A[half], A[half]);
            A[k] = kr;
            E = E * (1.0f - kr * kr);
        }

        // ---- Outputs: K (nsys floats) then a (nsys x 32 floats). ----------
        const long long s = blockBase + tid;
        out[s] = sqrtf(E);

        float* ab = out + (long long)nsys + s * (long long)M_ORDER;
        if (((nsys & 3) == 0)) {  // 16B-aligned vector path (true here)
            float4* ao = (float4*)ab;
#pragma unroll
            for (int q = 0; q < M_ORDER / 4; ++q)
                ao[q] = make_float4(A[4 * q + 0], A[4 * q + 1],
                                    A[4 * q + 2], A[4 * q + 3]);
        } else {
#pragma unroll
            for (int j = 0; j < M_ORDER; ++j) ab[j] = A[j];
        }
    }
}

extern "C" void kernel_launch(void* const* d_in, const int* in_sizes, int n_in,
                              void* d_out, int out_size, void* d_ws, size_t ws_size,
                              hipStream_t stream) {
    (void)n_in; (void)out_size; (void)d_ws; (void)ws_size;
    const float* r = (const float*)d_in[0];
    float* out = (float*)d_out;

    const int total = in_sizes[0];     // B*T*(M+1) = 64*1024*33
    const int nsys  = total / RLEN;    // 65536 systems

    const int grid = (nsys + SYS_PER_BLOCK - 1) / SYS_PER_BLOCK;  // 512
    pseudo_levinson_kernel<<<grid, SYS_PER_BLOCK, 0, stream>>>(r, out, nsys);
}